// ClusterNorm1d_37151467110657
// MI455X (gfx1250) — compile-verified
//
#include <hip/hip_runtime.h>
#include <hip/hip_bf16.h>

typedef __attribute__((ext_vector_type(2))) float v2f;
typedef __attribute__((ext_vector_type(8))) float v8f;

#define Dc    64      // cluster dim
#define Bc    4096    // batch
#define Kc    256     // clusters
#define CH    256     // staged column chunk
#define PITCH 257     // LDS row pitch (floats), breaks bank conflicts
#define GP    65      // 64x64 matrix pitch in LDS

// ---------------------------------------------------------------------------
// Generic tiled transpose over D slices: src[NI, D, NJ] -> dst[NJ, D, NI]
// Optionally accumulates per-(d, j) column sums (used to get the mean).
// ---------------------------------------------------------------------------
__global__ __launch_bounds__(256) void transpose_slices(
    const float* __restrict__ src, float* __restrict__ dst,
    int NI, int NJ, float* __restrict__ sumbuf, int accumulate)
{
    __shared__ float tile[32][33];
    const int d  = blockIdx.z;
    const int i0 = blockIdx.x * 32;
    const int j0 = blockIdx.y * 32;
    const int tx = threadIdx.x;      // 0..31
    const int ty = threadIdx.y;      // 0..7

    for (int r = 0; r < 4; ++r) {
        int i = i0 + ty + r * 8;
        tile[ty + r * 8][tx] = src[(size_t)i * Dc * NJ + (size_t)d * NJ + (j0 + tx)];
    }
    __syncthreads();
    if (accumulate && ty == 0) {
        float s = 0.f;
        #pragma unroll
        for (int r = 0; r < 32; ++r) s += tile[r][tx];
        atomicAdd(&sumbuf[d * NJ + (j0 + tx)], s);
    }
    for (int r = 0; r < 4; ++r) {
        int j = j0 + ty + r * 8;
        dst[(size_t)j * Dc * NI + (size_t)d * NI + (i0 + tx)] = tile[tx][ty + r * 8];
    }
}

// ---------------------------------------------------------------------------
// Per-cluster: Gram via WMMA f32, covariance finalize, Cholesky, L^-1, c=Linv*mu
// One workgroup (256 threads = 8 waves) per cluster.
// ---------------------------------------------------------------------------
__global__ __launch_bounds__(256) void gram_chol_kernel(
    const float* __restrict__ xT, const float* __restrict__ sumbuf,
    float* __restrict__ Linv, float* __restrict__ cvec)
{
    __shared__ float stg[Dc * PITCH];   // 64 x 256 staged chunk (padded)
    __shared__ float G[Dc * GP];        // 64 x 64 Gram/cov/L (padded)
    __shared__ float muL[Dc];

    const int k    = blockIdx.x;
    const int t    = threadIdx.x;
    const int lane = t & 31;
    const int wave = t >> 5;
    const int m    = lane & 15;
    const int half = lane >> 4;
    const float* xk = xT + (size_t)k * Dc * Bc;

    if (t < Dc) muL[t] = sumbuf[t * Kc + k] * (1.0f / (float)Bc);

    // each wave owns 2 of the 16 (d-tile, e-tile) output tiles
    const int p0 = wave * 2, p1 = wave * 2 + 1;
    const int ti0 = p0 >> 2, tj0 = p0 & 3;
    const int ti1 = p1 >> 2, tj1 = p1 & 3;
    v8f acc0 = {}; v8f acc1 = {};

    for (int b0 = 0; b0 < Bc; b0 += CH) {
        __syncthreads();                       // protect stg reuse
        // stage [64][CH] chunk, coalesced float4 loads
        #pragma unroll
        for (int it = 0; it < (Dc * CH / 4) / 256; ++it) {
            int lin = t + 256 * it;
            int row = lin >> 6;                // CH/4 = 64 float4 per row
            int c4  = (lin & 63) << 2;
            float4 v = *(const float4*)(xk + (size_t)row * Bc + b0 + c4);
            float* ds = &stg[row * PITCH + c4];
            ds[0] = v.x; ds[1] = v.y; ds[2] = v.z; ds[3] = v.w;
        }
        __syncthreads();

        for (int bb = 0; bb < CH; bb += 4) {
            int off = bb + 2 * half;
            v2f a0, b0f, a1, b1f;
            const float* r;
            r = &stg[(ti0 * 16 + m) * PITCH + off]; a0[0]  = r[0]; a0[1]  = r[1];
            r = &stg[(tj0 * 16 + m) * PITCH + off]; b0f[0] = r[0]; b0f[1] = r[1];
            r = &stg[(ti1 * 16 + m) * PITCH + off]; a1[0]  = r[0]; a1[1]  = r[1];
            r = &stg[(tj1 * 16 + m) * PITCH + off]; b1f[0] = r[0]; b1f[1] = r[1];
            acc0 = __builtin_amdgcn_wmma_f32_16x16x4_f32(false, a0, false, b0f,
                                                         (short)0, acc0, false, false);
            acc1 = __builtin_amdgcn_wmma_f32_16x16x4_f32(false, a1, false, b1f,
                                                         (short)0, acc1, false, false);
        }
    }

    // dump accumulators into LDS G (layout: VGPR v -> M = v + 8*half, N = m)
    #pragma unroll
    for (int v = 0; v < 8; ++v) {
        G[(ti0 * 16 + v + 8 * half) * GP + (tj0 * 16 + m)] = acc0[v];
        G[(ti1 * 16 + v + 8 * half) * GP + (tj1 * 16 + m)] = acc1[v];
    }
    __syncthreads();

    // cov = (G - B*mu*mu^T)/(B-1) + eps*I
    for (int idx = t; idx < Dc * Dc; idx += 256) {
        int i = idx >> 6, j = idx & 63;
        float g = (G[i * GP + j] - (float)Bc * muL[i] * muL[j]) * (1.0f / (float)(Bc - 1));
        if (i == j) g += 1e-4f;
        G[i * GP + j] = g;
    }
    __syncthreads();

    // in-LDS right-looking Cholesky (lower)
    for (int j = 0; j < Dc; ++j) {
        if (t == 0) G[j * GP + j] = sqrtf(G[j * GP + j]);
        __syncthreads();
        if (t > j && t < Dc) G[t * GP + j] /= G[j * GP + j];
        __syncthreads();
        for (int idx = t; idx < Dc * Dc; idx += 256) {
            int i = idx >> 6, c = idx & 63;
            if (i > j && c > j && c <= i)
                G[i * GP + c] -= G[i * GP + j] * G[c * GP + j];
        }
        __syncthreads();
    }

    // triangular inverse: LI = L^-1 (column c per thread), reuse stg as LI[64][GP]
    float* LI = stg;
    if (t < Dc) {
        int c = t;
        for (int i = c; i < Dc; ++i) {
            float s = (i == c) ? 1.0f : 0.0f;
            for (int j2 = c; j2 < i; ++j2) s -= G[i * GP + j2] * LI[j2 * GP + c];
            LI[i * GP + c] = s / G[i * GP + i];
        }
    }
    __syncthreads();

    // write Linv (dense, zero upper) and c = Linv * mu
    float* Lg = Linv + (size_t)k * Dc * Dc;
    for (int idx = t; idx < Dc * Dc; idx += 256) {
        int i = idx >> 6, j = idx & 63;
        Lg[idx] = (j <= i) ? LI[i * GP + j] : 0.0f;
    }
    if (t < Dc) {
        float s = 0.f;
        for (int e = 0; e <= t; ++e) s += LI[t * GP + e] * muL[e];
        cvec[k * Dc + t] = s;
    }
}

// ---------------------------------------------------------------------------
// Whiten in place: xT[k] <- Linv[k] * xT[k] - c[k]   (per 256-column chunk)
// ---------------------------------------------------------------------------
__global__ __launch_bounds__(256) void whiten_kernel(
    float* __restrict__ xT, const float* __restrict__ Linv,
    const float* __restrict__ cvec)
{
    __shared__ float stg[Dc * PITCH];
    __shared__ float LL[Dc * GP];
    __shared__ float cL[Dc];

    const int k    = blockIdx.x;
    const int b0   = blockIdx.y * CH;
    const int t    = threadIdx.x;
    const int lane = t & 31;
    const int wave = t >> 5;
    const int m    = lane & 15;
    const int half = lane >> 4;

    float* xk = xT + (size_t)k * Dc * Bc;
    const float* Lg = Linv + (size_t)k * Dc * Dc;

    for (int idx = t; idx < Dc * Dc; idx += 256)
        LL[(idx >> 6) * GP + (idx & 63)] = Lg[idx];
    if (t < Dc) cL[t] = cvec[k * Dc + t];

    #pragma unroll
    for (int it = 0; it < (Dc * CH / 4) / 256; ++it) {
        int lin = t + 256 * it;
        int row = lin >> 6;
        int c4  = (lin & 63) << 2;
        float4 v = *(const float4*)(xk + (size_t)row * Bc + b0 + c4);
        float* ds = &stg[row * PITCH + c4];
        ds[0] = v.x; ds[1] = v.y; ds[2] = v.z; ds[3] = v.w;
    }
    __syncthreads();

    // 64 output tiles (4 d-tiles x 16 b-tiles); 8 per wave
    for (int p = wave * 8; p < wave * 8 + 8; ++p) {
        int ti = p & 3;     // d tile
        int tj = p >> 2;    // b tile
        v8f acc = {};
        for (int ee = 0; ee < Dc; ee += 4) {
            int off = ee + 2 * half;
            v2f a, b;
            a[0] = LL[(ti * 16 + m) * GP + off];
            a[1] = LL[(ti * 16 + m) * GP + off + 1];
            b[0] = stg[off * PITCH + tj * 16 + m];
            b[1] = stg[(off + 1) * PITCH + tj * 16 + m];
            acc = __builtin_amdgcn_wmma_f32_16x16x4_f32(false, a, false, b,
                                                        (short)0, acc, false, false);
        }
        #pragma unroll
        for (int v = 0; v < 8; ++v) {
            int d = ti * 16 + v + 8 * half;
            xk[(size_t)d * Bc + b0 + tj * 16 + m] = acc[v] - cL[d];
        }
    }
}

// ---------------------------------------------------------------------------
extern "C" void kernel_launch(void* const* d_in, const int* in_sizes, int n_in,
                              void* d_out, int out_size, void* d_ws, size_t ws_size,
                              hipStream_t stream)
{
    const float* x = (const float*)d_in[0];
    float* out = (float*)d_out;
    float* ws  = (float*)d_ws;

    float* xT   = ws;                               // K*D*B floats (268 MB)
    float* sum  = xT + (size_t)Kc * Dc * Bc;        // D*K floats
    float* Linv = sum + Dc * Kc;                    // K*D*D floats (4 MB)
    float* cv   = Linv + (size_t)Kc * Dc * Dc;      // K*D floats

    hipMemsetAsync(sum, 0, Dc * Kc * sizeof(float), stream);

    dim3 tb(32, 8);
    // [B,D,K] -> xT[K,D,B], fused mean partial sums
    transpose_slices<<<dim3(Bc / 32, Kc / 32, Dc), tb, 0, stream>>>(
        x, xT, Bc, Kc, sum, 1);
    // per-cluster Gram (WMMA f32) + Cholesky + inverse + c
    gram_chol_kernel<<<Kc, 256, 0, stream>>>(xT, sum, Linv, cv);
    // in-place whiten (WMMA f32)
    whiten_kernel<<<dim3(Kc, Bc / CH), 256, 0, stream>>>(xT, Linv, cv);
    // zT[K,D,B] -> out[B,D,K]
    transpose_slices<<<dim3(Kc / 32, Bc / 32, Dc), tb, 0, stream>>>(
        xT, out, Kc, Bc, nullptr, 0);
}